// SpikingNeuralNetwork_40518721470628
// MI455X (gfx1250) — compile-verified
//
#include <hip/hip_runtime.h>
#include <stdint.h>

// ---------------------------------------------------------------------------
// Hodgkin-Huxley SNN scan for MI455X (gfx1250, wave32).
// B=16, T=4096, N=256. One neuron per lane; 32-thread (1-wave) blocks so each
// of the 128 waves owns a SIMD and its transcendental unit. Input current is
// double-buffer staged into LDS with CDNA5 async global->LDS copies
// (ASYNCcnt / s_wait_asynccnt); all 5 output streams use non-temporal stores.
// ---------------------------------------------------------------------------

#define HH_LOG2E 1.44269504088896340736f

static constexpr int HH_B = 16;
static constexpr int HH_T = 4096;
static constexpr int HH_N = 256;
static constexpr int TILE = 64;            // timesteps per async LDS tile
static constexpr int NTILES = HH_T / TILE; // 64 tiles

typedef int v4i __attribute__((ext_vector_type(4)));
typedef __attribute__((address_space(1))) v4i* glob_v4i_p;
typedef __attribute__((address_space(3))) v4i* lds_v4i_p;

#if __has_builtin(__builtin_amdgcn_global_load_async_to_lds_b128)
#define HH_ASYNC_LDS 1
#else
#define HH_ASYNC_LDS 0
#endif

#if __has_builtin(__builtin_amdgcn_s_wait_asynccnt)
#define HH_WAIT_ASYNC(n) __builtin_amdgcn_s_wait_asynccnt(n)
#else
#define HH_WAIT_ASYNC(n) asm volatile("s_wait_asynccnt " #n ::: "memory")
#endif

// fast math -> v_exp_f32 / v_rcp_f32 (trans pipe), v_med3_f32 clamp
__device__ __forceinline__ float fexp2(float x) { return __builtin_amdgcn_exp2f(x); }
__device__ __forceinline__ float frcp(float x)  { return __builtin_amdgcn_rcpf(x); }
__device__ __forceinline__ float fclamp01(float x) {
  return __builtin_amdgcn_fmed3f(x, 0.0f, 1.0f);  // single v_med3_f32
}

// All six HH rate exponentials folded to fma + exp2 (constants pre-merged).
__device__ __forceinline__ void hh_rates(float v, float& am, float& bm, float& ah,
                                         float& bh, float& an, float& bn) {
  const float e_m = fexp2(__builtin_fmaf(v, -0.1f * HH_LOG2E, -4.0f * HH_LOG2E));
  am = 0.1f * (v + 40.0f) * frcp(1.0f - e_m);
  bm = 4.0f * fexp2(__builtin_fmaf(v, -HH_LOG2E / 18.0f, -65.0f * HH_LOG2E / 18.0f));
  ah = 0.07f * fexp2(__builtin_fmaf(v, -HH_LOG2E / 20.0f, -65.0f * HH_LOG2E / 20.0f));
  bh = frcp(1.0f + fexp2(__builtin_fmaf(v, -0.1f * HH_LOG2E, -3.5f * HH_LOG2E)));
  const float e_n = fexp2(__builtin_fmaf(v, -0.1f * HH_LOG2E, -5.5f * HH_LOG2E));
  an = 0.01f * (v + 55.0f) * frcp(1.0f - e_n);
  bn = 0.125f * fexp2(__builtin_fmaf(v, -HH_LOG2E / 80.0f, -65.0f * HH_LOG2E / 80.0f));
}

#if HH_ASYNC_LDS
// Stage one 64-step x 32-neuron input tile (8 KB) into LDS.
// Each async b128 op moves 32 lanes x 16 B = 512 B = 4 tile rows of 128 B.
// Lane l handles row (l>>3), 16B chunk (l&7) of each 4-row group.
__device__ __forceinline__ void hh_stage_tile(const float* gtile, float* ltile,
                                              int lane) {
  const uintptr_t gaddr = (uintptr_t)gtile +
      (uintptr_t)((lane >> 3) * (HH_N * 4) + (lane & 7) * 16);
  const uintptr_t laddr = (uintptr_t)ltile + (uintptr_t)(lane * 16);
  glob_v4i_p g1 = (glob_v4i_p)gaddr;
  lds_v4i_p  l3 = (lds_v4i_p)laddr;  // generic LDS addr low 32b == LDS offset
#pragma unroll
  for (int i = 0; i < TILE / 4; ++i) {
    // per op: +4 timesteps => global +4*HH_N*4 B = 256 v4i; LDS +512 B = 32 v4i
    __builtin_amdgcn_global_load_async_to_lds_b128(g1 + i * HH_N,
                                                   l3 + i * 32,
                                                   /*imm offset*/ 0, /*cpol*/ 0);
  }
}
#endif

__global__ __launch_bounds__(32) void hh_kernel(const float* __restrict__ in,
                                                float* __restrict__ out) {
#if HH_ASYNC_LDS
  __shared__ float sbuf[2][TILE * 32];  // 16 KB double buffer
#endif
  const int lane = (int)threadIdx.x;  // 0..31
  const int wid  = (int)blockIdx.x;   // 0..127
  const int b    = wid >> 3;          // batch
  const int nc   = wid & 7;           // 32-neuron chunk within N=256

  const size_t S    = (size_t)HH_B * HH_T * HH_N;
  const size_t base = (size_t)b * HH_T * HH_N + (size_t)nc * 32 + (size_t)lane;
  const float* gin    = in + base;                                   // + t*N per step
  const float* gtile0 = in + ((size_t)b * HH_T * HH_N + (size_t)nc * 32);
  (void)gtile0;

  float* o_spk = out;
  float* o_v   = out + S;
  float* o_m   = out + 2 * S;
  float* o_h   = out + 3 * S;
  float* o_n   = out + 4 * S;

  // steady-state init at v0 = -65 mV (matches reference init)
  float v = -65.0f, m, h, n;
  {
    float am, bm, ah, bh, an, bn;
    hh_rates(v, am, bm, ah, bh, an, bn);
    m = am * frcp(am + bm);
    h = ah * frcp(ah + bh);
    n = an * frcp(an + bn);
  }

#if HH_ASYNC_LDS
  hh_stage_tile(gtile0, &sbuf[0][0], lane);
#endif

  size_t oidx = base;
  for (int tt = 0; tt < NTILES; ++tt) {
    const int cur = tt & 1;
    (void)cur;
#if HH_ASYNC_LDS
    if (tt + 1 < NTILES) {
      hh_stage_tile(gtile0 + (size_t)(tt + 1) * TILE * HH_N, &sbuf[cur ^ 1][0], lane);
      HH_WAIT_ASYNC(16);  // in-order ASYNCcnt: <=16 outstanding => tile tt landed
    } else {
      HH_WAIT_ASYNC(0);
    }
#endif
    if (tt + 2 < NTILES)
      __builtin_prefetch(gin + (size_t)(tt + 2) * TILE * HH_N, 0, 0);  // global_prefetch_b8

#pragma unroll 4
    for (int k = 0; k < TILE; ++k) {
#if HH_ASYNC_LDS
      const float i_t = sbuf[cur][k * 32 + lane];  // ds_load (pairs into 2addr)
#else
      const float i_t =
          __builtin_nontemporal_load(gin + (size_t)(tt * TILE + k) * HH_N);
#endif
      float am, bm, ah, bh, an, bn;
      hh_rates(v, am, bm, ah, bh, an, bn);

      // membrane currents from old state (independent of the 6 exps -> ILP)
      const float m3   = m * m * m;
      const float nn   = n * n;
      const float n4   = nn * nn;
      const float i_na = 120.0f * m3 * h * (v - 50.0f);
      const float i_k  = 36.0f * n4 * (v + 77.0f);
      const float i_l  = 0.3f * (v + 54.4f);
      const float v_new = __builtin_fmaf(((i_t - i_na) - i_k) - i_l, 0.01f, v);

      // gate updates: x + (a - (a+b)*x)*dt, clamped to [0,1] via v_med3_f32
      const float m_new =
          fclamp01(__builtin_fmaf(__builtin_fmaf(-(am + bm), m, am), 0.01f, m));
      const float h_new =
          fclamp01(__builtin_fmaf(__builtin_fmaf(-(ah + bh), h, ah), 0.01f, h));
      const float n_new =
          fclamp01(__builtin_fmaf(__builtin_fmaf(-(an + bn), n, an), 0.01f, n));

      // upward zero crossing; spikes[:,0,:] forced to 0
      const float spike =
          (((tt | k) != 0) && (v_new >= 0.0f) && (v < 0.0f)) ? 1.0f : 0.0f;

      // write-once streams -> non-temporal (TH=NT) stores
      __builtin_nontemporal_store(spike, o_spk + oidx);
      __builtin_nontemporal_store(v_new, o_v + oidx);
      __builtin_nontemporal_store(m_new, o_m + oidx);
      __builtin_nontemporal_store(h_new, o_h + oidx);
      __builtin_nontemporal_store(n_new, o_n + oidx);

      v = v_new; m = m_new; h = h_new; n = n_new;
      oidx += HH_N;
    }
  }
}

extern "C" void kernel_launch(void* const* d_in, const int* in_sizes, int n_in,
                              void* d_out, int out_size, void* d_ws, size_t ws_size,
                              hipStream_t stream) {
  (void)in_sizes; (void)n_in; (void)out_size; (void)d_ws; (void)ws_size;
  const float* in = (const float*)d_in[0];
  float* out = (float*)d_out;
  // 4096 neurons, one per lane: 128 single-wave workgroups spread across WGPs.
  hh_kernel<<<dim3(HH_B * HH_N / 32), dim3(32), 0, stream>>>(in, out);
}